// GraphEncoder_17557826306100
// MI455X (gfx1250) — compile-verified
//
#include <hip/hip_runtime.h>
#include <math.h>

#define FDIM 256          // H * C
#define CDIM 64
#define HEADS 4
#define NEG_SLOPE 0.2f
#define EPS_BN 1e-5f
#define EPS_SM 1e-16f

typedef __attribute__((ext_vector_type(2))) float v2f;
typedef __attribute__((ext_vector_type(8))) float v8f;

// ---------------- helpers ----------------

__device__ __forceinline__ float atomicMaxF(float* addr, float val) {
    int* ai = (int*)addr;
    int old = __float_as_int(*addr);
    while (__int_as_float(old) < val) {
        int assumed = old;
        old = atomicCAS(ai, assumed, __float_as_int(val));
        if (old == assumed) break;
    }
    return __int_as_float(old);
}

// src/dst for edge i over [real edges | self loops]
__device__ __forceinline__ void edge_sd(const int* __restrict__ ei, int E, long i,
                                        int& s, int& d) {
    if (i < (long)E) { s = ei[i]; d = ei[(size_t)E + i]; }
    else             { s = d = (int)(i - (long)E); }
}

// ---------------- kernels ----------------

// h[n, 256] = X[n, K] @ W[K, 256]   via V_WMMA_F32_16X16X4_F32
// One wave per 16x16 output tile. 4 waves/block each take one 16-col tile.
// f32 A (16x4): lanes 0-15 -> M, vgpr0/1 = K0/K1; lanes 16-31 -> K2/K3.
// f32 B (4x16): row-striped across lanes, mirrored halves.
// f32 C/D (16x16): lane&15 = col, vgpr r = row r (+8 for upper lanes).
__global__ __launch_bounds__(128) void k_gemm_wmma(const float* __restrict__ X,
                                                   const float* __restrict__ W,
                                                   float* __restrict__ Hf,
                                                   int n, int K) {
    const int lane = threadIdx.x & 31;
    const int wave = threadIdx.x >> 5;
    const int row0 = blockIdx.x * 16;
    const int col0 = (blockIdx.y * 4 + wave) * 16;
    const int rl   = lane & 15;
    const int kh   = lane >> 4;                 // 0 or 1 (K half)
    int row = row0 + rl;
    if (row >= n) row = n - 1;                  // clamp loads; stores predicated
    const int col = col0 + rl;
    const float* __restrict__ xr = X + (size_t)row * K;

    v8f acc = {};
    for (int k = 0; k < K; k += 4) {
        v2f a, b;
        a.x = xr[k + kh * 2 + 0];
        a.y = xr[k + kh * 2 + 1];
        b.x = W[(size_t)(k + kh * 2 + 0) * FDIM + col];
        b.y = W[(size_t)(k + kh * 2 + 1) * FDIM + col];
        if ((k & 31) == 0) __builtin_prefetch(xr + k + 32, 0, 0);
        acc = __builtin_amdgcn_wmma_f32_16x16x4_f32(
            /*neg_a=*/false, a, /*neg_b=*/false, b,
            /*c_mod=*/(short)0, acc, /*reuse_a=*/false, /*reuse_b=*/false);
    }
#pragma unroll
    for (int r = 0; r < 8; ++r) {
        int orow = row0 + r + kh * 8;
        if (orow < n) Hf[(size_t)orow * FDIM + col] = acc[r];
    }
}

// al[n,h] = sum_c h[n,h,c]*as[h,c] ; ar analogous
__global__ void k_attn(const float* __restrict__ Hf, const float* __restrict__ as_,
                       const float* __restrict__ ad_, float* __restrict__ al_,
                       float* __restrict__ ar_, int n) {
    long idx = (long)blockIdx.x * blockDim.x + threadIdx.x;
    if (idx >= (long)n * HEADS) return;
    int  h    = (int)(idx & (HEADS - 1));
    long node = idx >> 2;
    const float* __restrict__ hp  = Hf  + node * FDIM + (size_t)h * CDIM;
    const float* __restrict__ asp = as_ + (size_t)h * CDIM;
    const float* __restrict__ adp = ad_ + (size_t)h * CDIM;
    float sl = 0.f, sr = 0.f;
#pragma unroll 8
    for (int c = 0; c < CDIM; ++c) { float v = hp[c]; sl += v * asp[c]; sr += v * adp[c]; }
    al_[idx] = sl;
    ar_[idx] = sr;
}

__global__ void k_fill(float* __restrict__ p, float v, long cnt) {
    long i = (long)blockIdx.x * blockDim.x + threadIdx.x;
    if (i < cnt) p[i] = v;
}

// pass 1: e = leaky_relu(al[src]+ar[dst]); store e; segment max into m[dst,h]
__global__ void k_edge_max(const int* __restrict__ ei, int E, long EP,
                           const float* __restrict__ al_, const float* __restrict__ ar_,
                           float* __restrict__ ebuf, float* __restrict__ m_) {
    long idx = (long)blockIdx.x * blockDim.x + threadIdx.x;
    if (idx >= EP * HEADS) return;
    int  h = (int)(idx & 3);
    long e = idx >> 2;
    int s, d; edge_sd(ei, E, e, s, d);
    float v = al_[(size_t)s * HEADS + h] + ar_[(size_t)d * HEADS + h];
    v = (v > 0.f) ? v : NEG_SLOPE * v;
    ebuf[idx] = v;
    atomicMaxF(&m_[(size_t)d * HEADS + h], v);
}

// pass 2: p = exp(e - m[dst]); store p; segment sum into den[dst,h]
__global__ void k_edge_exp(const int* __restrict__ ei, int E, long EP,
                           const float* __restrict__ m_, float* __restrict__ ebuf,
                           float* __restrict__ den_) {
    long idx = (long)blockIdx.x * blockDim.x + threadIdx.x;
    if (idx >= EP * HEADS) return;
    int  h = (int)(idx & 3);
    long e = idx >> 2;
    int s, d; edge_sd(ei, E, e, s, d);
    (void)s;
    float p = expf(ebuf[idx] - m_[(size_t)d * HEADS + h]);
    ebuf[idx] = p;
    atomicAdd(&den_[(size_t)d * HEADS + h], p);
}

// pass 3: agg[dst,h,c] += h[src,h,c] * p/(den[dst,h]+eps). One block per edge.
__global__ __launch_bounds__(256) void k_edge_msg(const int* __restrict__ ei, int E, long EP,
                                                  const float* __restrict__ ebuf,
                                                  const float* __restrict__ den_,
                                                  const float* __restrict__ Hf,
                                                  float* __restrict__ agg_) {
    long e = (long)blockIdx.x;
    if (e >= EP) return;
    int s, d; edge_sd(ei, E, e, s, d);
    int t = threadIdx.x;            // t = h*64 + c
    int h = t >> 6;
    float alpha = ebuf[e * HEADS + h] / (den_[(size_t)d * HEADS + h] + EPS_SM);
    atomicAdd(&agg_[(size_t)d * FDIM + t], Hf[(size_t)s * FDIM + t] * alpha);
}

// head mean + bias
__global__ void k_finalize(const float* __restrict__ agg_, const float* __restrict__ bias,
                           float* __restrict__ out, int n) {
    long idx = (long)blockIdx.x * blockDim.x + threadIdx.x;
    if (idx >= (long)n * CDIM) return;
    long node = idx >> 6;
    int  c    = (int)(idx & 63);
    const float* __restrict__ ap = agg_ + node * FDIM + c;
    float v = 0.25f * (ap[0] + ap[CDIM] + ap[2 * CDIM] + ap[3 * CDIM]);
    out[idx] = v + bias[c];
}

// BN pass 1: per-channel sum / sumsq via LDS partials + global atomics
__global__ __launch_bounds__(256) void k_bn_stats(const float* __restrict__ X,
                                                  float* __restrict__ sums, int n) {
    __shared__ float ssum[256];
    __shared__ float ssq[256];
    int t = threadIdx.x;
    int c = t & 63;
    float s = 0.f, q = 0.f;
    for (long i = (long)blockIdx.x * 4 + (t >> 6); i < n; i += (long)gridDim.x * 4) {
        float v = X[i * CDIM + c];
        s += v; q += v * v;
    }
    ssum[t] = s; ssq[t] = q;
    __syncthreads();
    if (t < 64) {
        s = ssum[t] + ssum[t + 64] + ssum[t + 128] + ssum[t + 192];
        q = ssq[t]  + ssq[t + 64]  + ssq[t + 128]  + ssq[t + 192];
        atomicAdd(&sums[c], s);
        atomicAdd(&sums[64 + c], q);
    }
}

// BN pass 2: normalize + affine + ReLU (in place)
__global__ void k_bn_apply(float* __restrict__ X, const float* __restrict__ sums,
                           const float* __restrict__ g, const float* __restrict__ be, int n) {
    long idx = (long)blockIdx.x * blockDim.x + threadIdx.x;
    if (idx >= (long)n * CDIM) return;
    int c = (int)(idx & 63);
    float inv = 1.0f / (float)n;
    float mu  = sums[c] * inv;
    float var = sums[64 + c] * inv - mu * mu;
    float y = (X[idx] - mu) * rsqrtf(var + EPS_BN) * g[c] + be[c];
    X[idx] = (y > 0.f) ? y : 0.f;
}

// ---------------- launch ----------------

extern "C" void kernel_launch(void* const* d_in, const int* in_sizes, int n_in,
                              void* d_out, int out_size, void* d_ws, size_t ws_size,
                              hipStream_t stream) {
    (void)n_in; (void)out_size;
    const float* x  = (const float*)d_in[0];
    const int*   ei = (const int*)d_in[1];
    const int n = in_sizes[0] / 256;        // IN_C = 256
    const int E = in_sizes[1] / 2;          // edge_index is [2, E]
    const long EP = (long)E + n;            // + self loops

    const float* Wl[3]  = { (const float*)d_in[2],  (const float*)d_in[6],  (const float*)d_in[10] };
    const float* asl[3] = { (const float*)d_in[3],  (const float*)d_in[7],  (const float*)d_in[11] };
    const float* adl[3] = { (const float*)d_in[4],  (const float*)d_in[8],  (const float*)d_in[12] };
    const float* bl[3]  = { (const float*)d_in[5],  (const float*)d_in[9],  (const float*)d_in[13] };
    const float* gl[2]  = { (const float*)d_in[14], (const float*)d_in[16] };
    const float* bel[2] = { (const float*)d_in[15], (const float*)d_in[17] };

    // workspace carve-up
    size_t off = 0;
    auto carve = [&](size_t bytes) -> char* {
        char* p = (char*)d_ws + off;
        off = (off + bytes + 255) & ~(size_t)255;
        return p;
    };
    float* hfeat = (float*)carve((size_t)n * FDIM * 4);    // post-GEMM features [N,256]
    float* agg   = (float*)carve((size_t)n * FDIM * 4);    // scatter accumulator [N,256]
    float* xbuf  = (float*)carve((size_t)n * CDIM * 4);    // layer activations [N,64]
    float* al    = (float*)carve((size_t)n * HEADS * 4);
    float* ar    = (float*)carve((size_t)n * HEADS * 4);
    float* mbuf  = (float*)carve((size_t)n * HEADS * 4);
    float* den   = (float*)carve((size_t)n * HEADS * 4);
    float* ebuf  = (float*)carve((size_t)EP * HEADS * 4);  // per-edge e / p
    float* bns   = (float*)carve(128 * 4);                 // BN sum / sumsq
    if (off > ws_size) return;                             // insufficient scratch

    const long nh  = (long)n * HEADS;
    const long nc  = (long)n * CDIM;
    const long nf  = (long)n * FDIM;
    const long eh  = EP * HEADS;
    const dim3 blk(256);

    auto run_layer = [&](const float* xin, int K, int l, float* outp) {
        dim3 gg((n + 15) / 16, FDIM / 64);                 // 4 waves x 16-col tiles
        k_gemm_wmma<<<gg, 128, 0, stream>>>(xin, Wl[l], hfeat, n, K);
        k_attn<<<(unsigned)((nh + 255) / 256), blk, 0, stream>>>(hfeat, asl[l], adl[l], al, ar, n);
        k_fill<<<(unsigned)((nh + 255) / 256), blk, 0, stream>>>(mbuf, -INFINITY, nh);
        k_fill<<<(unsigned)((nh + 255) / 256), blk, 0, stream>>>(den, 0.f, nh);
        k_fill<<<(unsigned)((nf + 255) / 256), blk, 0, stream>>>(agg, 0.f, nf);
        k_edge_max<<<(unsigned)((eh + 255) / 256), blk, 0, stream>>>(ei, E, EP, al, ar, ebuf, mbuf);
        k_edge_exp<<<(unsigned)((eh + 255) / 256), blk, 0, stream>>>(ei, E, EP, mbuf, ebuf, den);
        k_edge_msg<<<(unsigned)EP, blk, 0, stream>>>(ei, E, EP, ebuf, den, hfeat, agg);
        k_finalize<<<(unsigned)((nc + 255) / 256), blk, 0, stream>>>(agg, bl[l], outp, n);
    };
    auto run_bn = [&](int l) {
        k_fill<<<1, 128, 0, stream>>>(bns, 0.f, 128);
        k_bn_stats<<<256, 256, 0, stream>>>(xbuf, bns, n);
        k_bn_apply<<<(unsigned)((nc + 255) / 256), blk, 0, stream>>>(xbuf, bns, gl[l], bel[l], n);
    };

    run_layer(x, 256, 0, xbuf);
    run_bn(0);
    run_layer(xbuf, CDIM, 1, xbuf);
    run_bn(1);
    run_layer(xbuf, CDIM, 2, (float*)d_out);
}